// Warping_Layer_53412213293360
// MI455X (gfx1250) — compile-verified
//
#include <hip/hip_runtime.h>

// dense_image_warp: out[b,y,x,c] = bilinear(image, (y,x) - flow[b,y,x])
// Shapes fixed by harness: B=8, H=1024, W=768, C=3, fp32.
#define BH   8
#define IH   1024
#define IW   768
#define IC   3
#define NPIX (BH * IH * IW)      // 6,291,456
#define TILE 256                 // pixels per block iteration (== blockDim.x)
#define NTILES (NPIX / TILE)     // 24,576 exactly

__global__ __launch_bounds__(256) void warp_kernel(const float* __restrict__ img,
                                                   const float* __restrict__ flow,
                                                   float* __restrict__ out)
{
    // Double-buffered LDS staging for the flow stream: 2 buffers * 256 lanes * 8B = 4 KB.
    // Written ONLY by global_load_async_to_lds_b64 (hardware), read by normal LDS loads.
    __shared__ float2 sflow[2 * TILE];

    const int tid = threadIdx.x;
    const unsigned long long flow_base = (unsigned long long)(uintptr_t)flow;

    const int tile0  = blockIdx.x;
    const int stride = gridDim.x;

    // Prime the pipeline: async-load tile0's flow into LDS buffer 0.
    if (tile0 < NTILES) {
        // Low 32 bits of the generic pointer into the LDS aperture == LDS byte
        // address (ISA flat-addressing rule). Passing it into the asm also
        // escapes sflow so the compiler keeps the ds_load readback.
        unsigned lds_addr = (unsigned)(uintptr_t)(&sflow[tid]);
        unsigned goff     = (unsigned)(tile0 * TILE + tid) * 8u;
        asm volatile("global_load_async_to_lds_b64 %0, %1, %2"
                     :: "v"(lds_addr), "v"(goff), "s"(flow_base)
                     : "memory");
    }

    int par = 0;
    for (int t = tile0; t < NTILES; t += stride, par ^= 1) {
        // Drain this wave's outstanding async copies: current tile's flow is in LDS.
        asm volatile("s_wait_asynccnt 0x0" ::: "memory");

        const int p = t * TILE + tid;
        const float2 f = sflow[par * TILE + tid];   // ds_load_b64

        // Kick off the NEXT tile's flow copy; it overlaps the gather+math below.
        const int tn = t + stride;
        if (tn < NTILES) {
            unsigned lds_addr = (unsigned)(uintptr_t)(&sflow[(par ^ 1) * TILE + tid]);
            unsigned goff     = (unsigned)(tn * TILE + tid) * 8u;
            asm volatile("global_load_async_to_lds_b64 %0, %1, %2"
                         :: "v"(lds_addr), "v"(goff), "s"(flow_base)
                         : "memory");
        }

        // p -> (b, y, x); W=768, H=1024 (power of two)
        const int x = p % IW;
        const int r = p / IW;
        const int y = r & (IH - 1);
        const int b = r >> 10;

        // query point = grid - flow ; flow[...,0]=dy, flow[...,1]=dx
        const float qy = (float)y - f.x;
        const float qx = (float)x - f.y;

        const float y0f = fminf(fmaxf(floorf(qy), 0.0f), (float)(IH - 2));
        const float x0f = fminf(fmaxf(floorf(qx), 0.0f), (float)(IW - 2));
        const float ay  = fminf(fmaxf(qy - y0f, 0.0f), 1.0f);
        const float ax  = fminf(fmaxf(qx - x0f, 0.0f), 1.0f);

        const int y0 = (int)y0f;
        const int x0 = (int)x0f;

        // top-left corner of the 2x2 neighborhood; rows are 6 contiguous floats
        const float* r0 = img + ((size_t)((b << 10) + y0) * IW + (size_t)x0) * IC;
        const float* r1 = r0 + (size_t)IW * IC;

        float o[IC];
#pragma unroll
        for (int c = 0; c < IC; ++c) {
            const float tl = r0[c];
            const float tr = r0[c + IC];
            const float bl = r1[c];
            const float br = r1[c + IC];
            const float top = fmaf(ax, tr - tl, tl);
            const float bot = fmaf(ax, br - bl, bl);
            o[c] = fmaf(ay, bot - top, top);
        }

        float* op = out + (size_t)p * IC;
        op[0] = o[0];
        op[1] = o[1];
        op[2] = o[2];
    }
}

extern "C" void kernel_launch(void* const* d_in, const int* in_sizes, int n_in,
                              void* d_out, int out_size, void* d_ws, size_t ws_size,
                              hipStream_t stream)
{
    (void)in_sizes; (void)n_in; (void)out_size; (void)d_ws; (void)ws_size;
    const float* img  = (const float*)d_in[0];   // [8,1024,768,3] f32
    const float* flow = (const float*)d_in[1];   // [8,1024,768,2] f32
    float* out = (float*)d_out;                  // [8,1024,768,3] f32

    // 4096 blocks * 256 threads (8 wave32/block); each block pipelines 6 tiles.
    warp_kernel<<<4096, 256, 0, stream>>>(img, flow, out);
}